// GMP_72602127171647
// MI455X (gfx1250) — compile-verified
//
#include <hip/hip_runtime.h>

// GMP forward for MI455X (gfx1250, wave32).
// out[b,j] = sum_i x[b,j+i-7] * ( W0[i] + F[b, j+i-7, i] )
// F[b,r,i] = sum_{q=0..23} P[b,r,q] * Wq[q,i]
//   with q = 3m+d, P[b,r,3m+d] = |x[b,r+m-7]|^(d+1), Wq[q,i] = Weight[i, 1+q]
// F is computed per 16-row tile with chained V_WMMA_F32_16X16X4_F32 (full fp32,
// matches the fp32 JAX reference; K=24 -> 6 chained K=4 WMMAs).

typedef float v2f __attribute__((ext_vector_type(2)));
typedef float v8f __attribute__((ext_vector_type(8)));

#define TJ 256   // outputs per block (8 waves x 32)

__global__ __launch_bounds__(256)
void gmp_wmma_f32_kernel(const float* __restrict__ x,
                         const float* __restrict__ Wt,
                         float* __restrict__ out, int T)
{
    // LDS: x tile + halo, interleaved amplitude powers, F (stride 9 to break
    // half-wave bank collisions), weights.  Total ~16.2 KB of the 320 KB WGP LDS.
    __shared__ float xre[272];         // x real, r in [j0-16, j0+255]
    __shared__ float xim[272];         // x imag
    __shared__ float Pint[288 * 3];    // a^(d+1) interleaved, u in [j0-32, j0+255]
    __shared__ float Flds[272 * 9];    // F[r - (j0-16)][i], padded stride 9
    __shared__ float Wlds[200];        // raw 8x25 Weight

    const int t  = threadIdx.x;
    const int j0 = blockIdx.x * TJ;
    const size_t base = (size_t)blockIdx.y * T;
    const float2* __restrict__ x2 = (const float2*)x;

    // ---------------- load phase ----------------
    for (int s = t; s < 272; s += 256) {
        int u = j0 - 16 + s;
        float re = 0.f, im = 0.f;
        if (u >= 0) { float2 v = x2[base + u]; re = v.x; im = v.y; }
        xre[s] = re; xim[s] = im;
    }
    for (int s = t; s < 288; s += 256) {
        int u = j0 - 32 + s;
        float a = 0.f;
        if (u >= 0) {
            float2 v = x2[base + u];
            a = sqrtf(fmaf(v.x, v.x, v.y * v.y));
        }
        Pint[s * 3 + 0] = a;           // a^1
        Pint[s * 3 + 1] = a * a;       // a^2
        Pint[s * 3 + 2] = a * a * a;   // a^3
    }
    if (t < 200) Wlds[t] = Wt[t];
    __syncthreads();

    // ---------------- WMMA phase: F[r,i] ----------------
    const int wave  = t >> 5;          // 0..7
    const int lane  = t & 31;
    const int row   = lane & 15;       // A: M row / B,D: N column
    const int khalf = lane >> 4;       // K-half select (A/B), M-half select (D)

    // B fragments (4x16, K = vgpr + 2*khalf per ISA 16x4-f32 layout mirrored):
    //   B[q, n] = (n < 8) ? Weight[n, 1+q] : 0
    v2f Bf[6];
#pragma unroll
    for (int step = 0; step < 6; ++step) {
        int q0 = 4 * step + 2 * khalf;
        float b0 = (row < 8) ? Wlds[row * 25 + 1 + q0]     : 0.f;
        float b1 = (row < 8) ? Wlds[row * 25 + 1 + q0 + 1] : 0.f;
        Bf[step].x = b0;
        Bf[step].y = b1;
    }

    // Wave w covers r in [j0 + 32w - 16, j0 + 32w + 31] : 3 tiles of 16 rows.
    // A[row, q] = P_d[r_row + m - 7] = Pint[3*(32w + 16*tau + row + 9) + q].
    for (int tau = 0; tau < 3; ++tau) {
        const int pbase = 3 * (32 * wave + 16 * tau + row + 9);
        v8f c = {};                    // accumulator starts at 0 (W0 added later)
#pragma unroll
        for (int step = 0; step < 6; ++step) {
            int q0 = 4 * step + 2 * khalf;
            v2f a;
            a.x = Pint[pbase + q0];
            a.y = Pint[pbase + q0 + 1];
            c = __builtin_amdgcn_wmma_f32_16x16x4_f32(
                    /*neg_a=*/false, a, /*neg_b=*/false, Bf[step],
                    /*c_mod=*/(short)0, c, /*reuse_a=*/false, /*reuse_b=*/false);
        }
        // D layout: vgpr v holds element (M = v + 8*khalf, N = row); keep N<8.
        if (row < 8) {
            const int frbase = 32 * wave + 16 * tau + 8 * khalf;
#pragma unroll
            for (int v = 0; v < 8; ++v) {
                Flds[(frbase + v) * 9 + row] = c[v];
            }
        }
    }
    __syncthreads();

    // ---------------- combine phase ----------------
    // out[j] = sum_i x[r]*(W0[i] + F[r,i]),  r = j+i-7, LDS row = t+i+9.
    float ar = 0.f, ai = 0.f;
#pragma unroll
    for (int i = 0; i < 8; ++i) {
        int ri = t + i + 9;
        float coef = Wlds[i * 25] + Flds[ri * 9 + i];
        ar = fmaf(xre[ri], coef, ar);
        ai = fmaf(xim[ri], coef, ai);
    }
    ((float2*)out)[base + j0 + t] = make_float2(ar, ai);
}

extern "C" void kernel_launch(void* const* d_in, const int* in_sizes, int n_in,
                              void* d_out, int out_size, void* d_ws, size_t ws_size,
                              hipStream_t stream) {
    const float* x  = (const float*)d_in[0];   // (B, T, 2) f32
    // d_in[1] = h_0 (unused by the reference module)
    const float* Wt = (const float*)d_in[2];   // (8, 25) f32
    float* out = (float*)d_out;                // (B, T, 2) f32

    const int T = 16384;
    const int B = in_sizes[0] / (2 * T);

    dim3 grid(T / TJ, B);
    gmp_wmma_f32_kernel<<<grid, 256, 0, stream>>>(x, Wt, out, T);
}